// ParallelHybridScan_61065845015120
// MI455X (gfx1250) — compile-verified
//
#include <hip/hip_runtime.h>

// Problem constants (from reference)
#define B_SZ 4
#define L_SZ 2048
#define H_SZ 16
#define D_SZ 64
#define N_SZ 64
#define LD 68                 // padded LDS stride for *x64 tiles (16B aligned, conflict-free)

// ---- single-pass (fallback) config ----
#define CT 16
#define CHUNKS (L_SZ / CT)    // 128
#define NT 128
#define LDS_S 68
#define LDP 20

// ---- chunk-parallel (3-phase) config ----
#define CT2 64                // chunk length
#define CH2 (L_SZ / CT2)      // 32 chunks
#define BH (B_SZ * H_SZ)      // 64

typedef __attribute__((ext_vector_type(2))) float v2f;
typedef __attribute__((ext_vector_type(8))) float v8f;

__device__ __forceinline__ v8f wmma_f32(v2f a, v2f b, v8f c) {
  // D = A(16x4) * B(4x16) + C(16x16), fp32, wave32
  return __builtin_amdgcn_wmma_f32_16x16x4_f32(false, a, false, b, (short)0, c,
                                               false, false);
}

// ---- CDNA5 async global->LDS DMA (ASYNCcnt-tracked), 16B per lane ----------
__device__ __forceinline__ void async_ld_b128(void* lds_ptr, const void* gptr) {
  unsigned lds_addr = (unsigned)(uintptr_t)lds_ptr;  // flat->LDS uses addr[31:0]
  asm volatile("global_load_async_to_lds_b128 %0, %1, off"
               :: "v"(lds_addr), "v"(gptr)
               : "memory");
}
__device__ __forceinline__ void wait_asynccnt0() {
  asm volatile("s_wait_asynccnt 0" ::: "memory");
}

// ---- WMMA fragment loaders (fp32 16x16x4) -------------------------------
// A 16x4: lanes 0-15 hold M=lane with K=0(v0),K=1(v1); lanes 16-31 M=lane-16, K=2,3
__device__ __forceinline__ v2f lda_row(const float* M, int ld, int r0, int c0, int lane) {
  int m = lane & 15, k2 = (lane >> 4) << 1;
  v2f a; a.x = M[(r0 + m) * ld + c0 + k2]; a.y = M[(r0 + m) * ld + c0 + k2 + 1];
  return a;
}
// A from transposed storage: logical A[m][k] = M[(c0+k)*ld + r0+m]
__device__ __forceinline__ v2f lda_col(const float* M, int ld, int r0, int c0, int lane) {
  int m = lane & 15, k2 = (lane >> 4) << 1;
  v2f a; a.x = M[(c0 + k2) * ld + r0 + m]; a.y = M[(c0 + k2 + 1) * ld + r0 + m];
  return a;
}
// B 4x16: lanes 0-15 hold N=lane with K=0(v0),K=1(v1); lanes 16-31 N=lane-16, K=2,3
__device__ __forceinline__ v2f ldb_row(const float* M, int ld, int r0, int c0, int lane) {
  int n = lane & 15, k2 = (lane >> 4) << 1;
  v2f b; b.x = M[(r0 + k2) * ld + c0 + n]; b.y = M[(r0 + k2 + 1) * ld + c0 + n];
  return b;
}
// B from transposed storage: logical B[k][n] = M[(c0+n)*ld + r0+k]
__device__ __forceinline__ v2f ldb_col(const float* M, int ld, int r0, int c0, int lane) {
  int n = lane & 15, k2 = (lane >> 4) << 1;
  v2f b; b.x = M[(c0 + n) * ld + r0 + k2]; b.y = M[(c0 + n) * ld + r0 + k2 + 1];
  return b;
}

// ===========================================================================
// Phase 1: per (bh, chunk) compute dS = Kc^T @ V  and Dend, into workspace.
// ===========================================================================
__global__ __launch_bounds__(128)
void p1_chunk_state(const float* __restrict__ kp, const float* __restrict__ vp,
                    const float* __restrict__ dtp, const float* __restrict__ bvp,
                    const float* __restrict__ bdp,
                    float* __restrict__ dS, float* __restrict__ dEnd) {
  __shared__ float sK[CT2 * LD], sV[CT2 * LD], sDt[CT2 * LD], sKc[CT2 * LD];
  __shared__ float sA[CT2], sEend[D_SZ], sBD[D_SZ];

  const int blk = blockIdx.x;
  const int bh = blk / CH2, c = blk % CH2;
  const int b = bh >> 4, h = bh & 15;
  const int tid = threadIdx.x;
  const int wave = tid >> 5, lane = tid & 31;
  const int hh = lane >> 4, nn = lane & 15;
  const int rowstride = H_SZ * D_SZ;
  const int t0 = c * CT2;
  const size_t gbase = ((size_t)(b * L_SZ + t0) * H_SZ + h) * (size_t)D_SZ;
  const size_t bbase = (size_t)(b * L_SZ + t0) * (size_t)N_SZ;
  const size_t slot = ((size_t)bh * CH2 + c) * (size_t)(D_SZ * D_SZ);

  if (tid < D_SZ) sBD[tid] = bdp[h * D_SZ + tid];
  // async DMA staging: global -> LDS, no VGPR round trip
  for (int i = tid; i < CT2 * 16; i += 128) {
    int t = i >> 4, q = (i & 15) * 4;
    size_t go = gbase + (size_t)t * rowstride + q;
    async_ld_b128(&sK[t * LD + q],  kp + go);
    async_ld_b128(&sV[t * LD + q],  vp + go);
    async_ld_b128(&sDt[t * LD + q], dtp + go);
  }
  wait_asynccnt0();
  __syncthreads();

  if (tid < D_SZ) {                 // cumsum of bd*dt per key dim, in place
    float bd = sBD[tid];
    float run = 0.0f;
    for (int t = 0; t < CT2; ++t) {
      run += sDt[t * LD + tid];
      sDt[t * LD + tid] = bd * run;
    }
    float de = expf(-bd * run);     // Dend[k]
    sEend[tid] = de;
    dEnd[((size_t)bh * CH2 + c) * D_SZ + tid] = de;
  } else {                          // a_t = mean(B_t) over N
    int t = tid - D_SZ;             // 0..63
    float s = 0.0f;
    for (int n = 0; n < N_SZ; ++n) s += bvp[bbase + (size_t)t * N_SZ + n];
    sA[t] = s * (1.0f / (float)N_SZ);
  }
  __syncthreads();

  for (int i = tid; i < CT2 * D_SZ; i += 128) {   // Kc = a*k*e^{-(cumEnd-cum)}
    int t = i >> 6, kd = i & 63;
    float cum = sDt[t * LD + kd];
    sKc[t * LD + kd] = sA[t] * sK[t * LD + kd] * (sEend[kd] * expf(cum));
  }
  __syncthreads();

  // dS tiles: wave owns k-rows [16w, 16w+16); 4 v-blocks each; K over time = 64
  for (int cb = 0; cb < 4; ++cb) {
    v8f acc = {0.f, 0.f, 0.f, 0.f, 0.f, 0.f, 0.f, 0.f};
#pragma unroll
    for (int i = 0; i < CT2 / 4; ++i) {
      v2f a = lda_col(sKc, LD, 16 * wave, 4 * i, lane);   // A[m][s] = Kc[s][16w+m]
      v2f bb = ldb_row(sV, LD, 4 * i, 16 * cb, lane);
      acc = wmma_f32(a, bb, acc);
    }
#pragma unroll
    for (int i = 0; i < 8; ++i) {
      int row = 16 * wave + i + 8 * hh;
      dS[slot + (size_t)row * D_SZ + 16 * cb + nn] = acc[i];
    }
  }
}

// ===========================================================================
// Phase 2: per (bh) scan over chunks: write start-states in place, final
// state to d_out tail. Serial depth = 32 FMAs per element, loads independent.
// ===========================================================================
__global__ __launch_bounds__(256)
void p2_scan(float* __restrict__ dS, const float* __restrict__ dEnd,
             float* __restrict__ outp) {
  const int bh = blockIdx.x;
  const int tid = threadIdx.x;
  const size_t eb = (size_t)tid * 16;   // 16 consecutive state elements
  const int kd = tid >> 2;              // all 16 share one key-dim row
  float s[16];
#pragma unroll
  for (int j = 0; j < 16; ++j) s[j] = 0.0f;

#pragma unroll 4
  for (int c = 0; c < CH2; ++c) {
    size_t slot = ((size_t)bh * CH2 + c) * (size_t)(D_SZ * D_SZ) + eb;
    float d[16];
    *(float4*)&d[0]  = *(const float4*)&dS[slot];
    *(float4*)&d[4]  = *(const float4*)&dS[slot + 4];
    *(float4*)&d[8]  = *(const float4*)&dS[slot + 8];
    *(float4*)&d[12] = *(const float4*)&dS[slot + 12];
    float de = dEnd[((size_t)bh * CH2 + c) * D_SZ + kd];
    // overwrite the slot with the chunk's START state (read before write)
    *(float4*)&dS[slot]      = *(const float4*)&s[0];
    *(float4*)&dS[slot + 4]  = *(const float4*)&s[4];
    *(float4*)&dS[slot + 8]  = *(const float4*)&s[8];
    *(float4*)&dS[slot + 12] = *(const float4*)&s[12];
#pragma unroll
    for (int j = 0; j < 16; ++j) s[j] = de * s[j] + d[j];
  }
  // final state -> d_out tail ([B,H,D,D] == bh*4096 + k*64 + v)
  const size_t OUT0 = (size_t)B_SZ * L_SZ * H_SZ * D_SZ;
  float* so = outp + OUT0 + (size_t)bh * (D_SZ * D_SZ) + eb;
  *(float4*)&so[0]  = *(const float4*)&s[0];
  *(float4*)&so[4]  = *(const float4*)&s[4];
  *(float4*)&so[8]  = *(const float4*)&s[8];
  *(float4*)&so[12] = *(const float4*)&s[12];
}

// ===========================================================================
// Phase 3: per (bh, chunk): O = Kt @ S_start + tril(Kt@Kh^T) @ V, epilogue.
// Each wave computes TWO adjacent v-tiles sharing every A fragment.
// ===========================================================================
__global__ __launch_bounds__(256)
void p3_output(const float* __restrict__ kp, const float* __restrict__ vp,
               const float* __restrict__ dtp, const float* __restrict__ bvp,
               const float* __restrict__ rp, const float* __restrict__ gp,
               const float* __restrict__ bdp, const float* __restrict__ tfp,
               const float* __restrict__ dS, float* __restrict__ outp) {
  __shared__ float sK[CT2 * LD], sV[CT2 * LD], sDt[CT2 * LD];
  __shared__ float sKt[CT2 * LD], sKh[CT2 * LD];
  __shared__ float sS[D_SZ * LD], sP[CT2 * LD];
  __shared__ float sA[CT2], sTF[D_SZ], sBD[D_SZ];

  const int blk = blockIdx.x;
  const int bh = blk / CH2, c = blk % CH2;
  const int b = bh >> 4, h = bh & 15;
  const int tid = threadIdx.x;
  const int wave = tid >> 5, lane = tid & 31;
  const int hh = lane >> 4, nn = lane & 15;
  const int rowstride = H_SZ * D_SZ;
  const int t0 = c * CT2;
  const size_t gbase = ((size_t)(b * L_SZ + t0) * H_SZ + h) * (size_t)D_SZ;
  const size_t bbase = (size_t)(b * L_SZ + t0) * (size_t)N_SZ;
  const size_t slot = ((size_t)bh * CH2 + c) * (size_t)(D_SZ * D_SZ);

  if (tid < D_SZ) {
    sTF[tid] = tfp[h * D_SZ + tid];
    sBD[tid] = bdp[h * D_SZ + tid];
  }
  // async DMA staging: k, v, dt and the chunk's start-state
  for (int i = tid; i < CT2 * 16; i += 256) {
    int t = i >> 4, q = (i & 15) * 4;
    size_t go = gbase + (size_t)t * rowstride + q;
    async_ld_b128(&sK[t * LD + q],  kp + go);
    async_ld_b128(&sV[t * LD + q],  vp + go);
    async_ld_b128(&sDt[t * LD + q], dtp + go);
    async_ld_b128(&sS[t * LD + q],  dS + slot + (size_t)t * D_SZ + q);
  }
  // stream r/g toward caches for the epilogue (global_prefetch_b8)
  __builtin_prefetch(rp + gbase + (size_t)tid * 16, 0, 0);
  __builtin_prefetch(gp + gbase + (size_t)tid * 16, 0, 0);
  wait_asynccnt0();
  __syncthreads();

  if (tid < D_SZ) {                 // cumsum of bd*dt per key dim, in place
    float bd = sBD[tid];
    float run = 0.0f;
    for (int t = 0; t < CT2; ++t) {
      run += sDt[t * LD + tid];
      sDt[t * LD + tid] = bd * run;
    }
  } else if (tid < D_SZ + CT2) {    // a_t = mean(B_t)
    int t = tid - D_SZ;
    float s = 0.0f;
    for (int n = 0; n < N_SZ; ++n) s += bvp[bbase + (size_t)t * N_SZ + n];
    sA[t] = s * (1.0f / (float)N_SZ);
  }
  __syncthreads();

  for (int i = tid; i < CT2 * D_SZ; i += 256) {   // Kt = k*e^{-cum}, Kh = a*k*e^{+cum}
    int t = i >> 6, kd = i & 63;
    float cum = sDt[t * LD + kd];
    float em = expf(-cum);
    float kv = sK[t * LD + kd];
    sKt[t * LD + kd] = kv * em;
    sKh[t * LD + kd] = sA[t] * kv * (1.0f / em);
  }
  __syncthreads();

  // ---- P = tril(Kt @ Kh^T): only lower-triangular 16x16 tile blocks ----
  const int PTI[10] = {0, 1, 1, 2, 2, 2, 3, 3, 3, 3};
  const int PSI[10] = {0, 0, 1, 0, 1, 2, 0, 1, 2, 3};
  for (int idx = wave; idx < 10; idx += 8) {
    int ti2 = PTI[idx], si = PSI[idx];
    v8f acc = {0.f, 0.f, 0.f, 0.f, 0.f, 0.f, 0.f, 0.f};
#pragma unroll
    for (int i = 0; i < 16; ++i) {                       // K over kdim = 64
      v2f a = lda_row(sKt, LD, 16 * ti2, 4 * i, lane);
      v2f bb = ldb_col(sKh, LD, 4 * i, 16 * si, lane);   // B[kd][s] = Kh[16si+s][kd]
      acc = wmma_f32(a, bb, acc);
    }
    bool diag = (ti2 == si);
#pragma unroll
    for (int i = 0; i < 8; ++i) {
      int m = i + 8 * hh;
      float pv = acc[i];
      if (diag && nn > m) pv = 0.0f;                     // causal mask s <= t
      sP[(16 * ti2 + m) * LD + 16 * si + nn] = pv;
    }
  }
  __syncthreads();

  // ---- O tiles: wave -> (ti, {vi0, vi0+1}); A fragments shared, 2 chains ----
  const int ti = wave & 3, vi0 = (wave >> 2) * 2;
  v8f O0 = {0.f, 0.f, 0.f, 0.f, 0.f, 0.f, 0.f, 0.f};
  v8f O1 = {0.f, 0.f, 0.f, 0.f, 0.f, 0.f, 0.f, 0.f};
#pragma unroll
  for (int i = 0; i < 16; ++i) {                         // inter: K over kdim
    v2f a  = lda_row(sKt, LD, 16 * ti, 4 * i, lane);
    v2f b0 = ldb_row(sS, LD, 4 * i, 16 * vi0, lane);
    v2f b1 = ldb_row(sS, LD, 4 * i, 16 * vi0 + 16, lane);
    O0 = wmma_f32(a, b0, O0);
    O1 = wmma_f32(a, b1, O1);
  }
  const int ks = 4 * (ti + 1);                           // intra: only s-tiles <= ti
  for (int i = 0; i < ks; ++i) {
    v2f a  = lda_row(sP, LD, 16 * ti, 4 * i, lane);
    v2f b0 = ldb_row(sV, LD, 4 * i, 16 * vi0, lane);
    v2f b1 = ldb_row(sV, LD, 4 * i, 16 * vi0 + 16, lane);
    O0 = wmma_f32(a, b0, O0);
    O1 = wmma_f32(a, b1, O1);
  }
  // ---- epilogue: out = r * (O + tf*k*v) * g, both tiles ----
#pragma unroll
  for (int half = 0; half < 2; ++half) {
    int vcol = 16 * (vi0 + half) + nn;
    float tf = sTF[vcol];
#pragma unroll
    for (int i = 0; i < 8; ++i) {
      int t = 16 * ti + i + 8 * hh;
      size_t go = gbase + (size_t)t * rowstride + vcol;
      float ov = half ? O1[i] : O0[i];
      float wkv = ov + tf * sK[t * LD + vcol] * sV[t * LD + vcol];
      outp[go] = rp[go] * wkv * gp[go];
    }
  }
}

// ===========================================================================
// Fallback: single-pass chunked scan (round 1), used if ws too small
// ===========================================================================
__global__ __launch_bounds__(NT)
void gla_chunk_scan_kernel(const float* __restrict__ kp, const float* __restrict__ vp,
                           const float* __restrict__ dtp, const float* __restrict__ bvp,
                           const float* __restrict__ rp, const float* __restrict__ gp,
                           const float* __restrict__ bdp, const float* __restrict__ tfp,
                           float* __restrict__ outp) {
  __shared__ float sK[CT * LD], sV[CT * LD], sDt[CT * LD], sB[CT * LD];
  __shared__ float sKt[CT * LD], sKh[CT * LD], sKc[CT * LD];
  __shared__ float sS[D_SZ * LDS_S];
  __shared__ float sP[CT * LDP];
  __shared__ float sA[CT], sEend[D_SZ], sTF[D_SZ], sBD[D_SZ];

  const int bh = blockIdx.x;
  const int b = bh / H_SZ, h = bh % H_SZ;
  const int tid = threadIdx.x;
  const int wave = tid >> 5, lane = tid & 31;
  const int hh = lane >> 4, nn = lane & 15;
  const int rowstride = H_SZ * D_SZ;

  for (int i = tid; i < D_SZ * LDS_S; i += NT) sS[i] = 0.0f;
  if (tid < D_SZ) {
    sTF[tid] = tfp[h * D_SZ + tid];
    sBD[tid] = bdp[h * D_SZ + tid];
  }
  __syncthreads();

  for (int c = 0; c < CHUNKS; ++c) {
    const int t0 = c * CT;
    const size_t gbase = ((size_t)(b * L_SZ + t0) * H_SZ + h) * (size_t)D_SZ;
    const size_t bbase = (size_t)(b * L_SZ + t0) * (size_t)N_SZ;

    for (int i = tid; i < CT * 16; i += NT) {
      int t = i >> 4, q = (i & 15) * 4;
      size_t go = gbase + (size_t)t * rowstride + q;
      *(float4*)&sK[t * LD + q]  = *(const float4*)&kp[go];
      *(float4*)&sV[t * LD + q]  = *(const float4*)&vp[go];
      *(float4*)&sDt[t * LD + q] = *(const float4*)&dtp[go];
      *(float4*)&sB[t * LD + q]  = *(const float4*)&bvp[bbase + (size_t)t * N_SZ + q];
    }
    if (c + 1 < CHUNKS) {
      __builtin_prefetch(kp + gbase + (size_t)CT * rowstride + (size_t)tid * 8, 0, 0);
      __builtin_prefetch(vp + gbase + (size_t)CT * rowstride + (size_t)tid * 8, 0, 0);
    }
    __syncthreads();

    if (tid < D_SZ) {
      float bd = sBD[tid];
      float run = 0.0f;
      for (int t = 0; t < CT; ++t) {
        run += sDt[t * LD + tid];
        sDt[t * LD + tid] = bd * run;
      }
      sEend[tid] = expf(-bd * run);
    } else if (tid < D_SZ + CT) {
      int t = tid - D_SZ;
      float s = 0.0f;
      for (int n = 0; n < N_SZ; ++n) s += sB[t * LD + n];
      sA[t] = s * (1.0f / (float)N_SZ);
    }
    __syncthreads();

    for (int i = tid; i < CT * D_SZ; i += NT) {
      int t = i >> 6, kd = i & 63;
      float cum = sDt[t * LD + kd];
      float em = expf(-cum);
      float kv = sK[t * LD + kd];
      float a = sA[t];
      float inv = 1.0f / em;
      sKt[t * LD + kd] = kv * em;
      sKh[t * LD + kd] = a * kv * inv;
      sKc[t * LD + kd] = a * kv * sEend[kd] * inv;
    }
    __syncthreads();

    v8f O = {0.f, 0.f, 0.f, 0.f, 0.f, 0.f, 0.f, 0.f};
#pragma unroll
    for (int i = 0; i < 16; ++i) {
      v2f a = lda_row(sKt, LD, 0, 4 * i, lane);
      v2f bb = ldb_row(sS, LDS_S, 4 * i, 16 * wave, lane);
      O = wmma_f32(a, bb, O);
    }

    if (wave == 0) {
      v8f P = {0.f, 0.f, 0.f, 0.f, 0.f, 0.f, 0.f, 0.f};
#pragma unroll
      for (int i = 0; i < 16; ++i) {
        v2f a = lda_row(sKt, LD, 0, 4 * i, lane);
        v2f bb = ldb_col(sKh, LD, 4 * i, 0, lane);
        P = wmma_f32(a, bb, P);
      }
#pragma unroll
      for (int i = 0; i < 8; ++i) {
        int m = i + 8 * hh;
        sP[m * LDP + nn] = (nn <= m) ? P[i] : 0.0f;
      }
    }
    __syncthreads();

#pragma unroll
    for (int i = 0; i < 4; ++i) {
      v2f a = lda_row(sP, LDP, 0, 4 * i, lane);
      v2f bb = ldb_row(sV, LD, 4 * i, 16 * wave, lane);
      O = wmma_f32(a, bb, O);
    }

#pragma unroll
    for (int cb = 0; cb < 4; ++cb) {
      v8f Sacc = {0.f, 0.f, 0.f, 0.f, 0.f, 0.f, 0.f, 0.f};
#pragma unroll
      for (int i = 0; i < 4; ++i) {
        v2f a = lda_col(sKc, LD, 16 * wave, 4 * i, lane);
        v2f bb = ldb_row(sV, LD, 4 * i, 16 * cb, lane);
        Sacc = wmma_f32(a, bb, Sacc);
      }
#pragma unroll
      for (int i = 0; i < 8; ++i) {
        int row = 16 * wave + i + 8 * hh;
        int col = 16 * cb + nn;
        float sOld = sS[row * LDS_S + col];
        sS[row * LDS_S + col] = sEend[row] * sOld + Sacc[i];
      }
    }

    {
      int vcol = 16 * wave + nn;
      float tf = sTF[vcol];
#pragma unroll
      for (int i = 0; i < 8; ++i) {
        int t = i + 8 * hh;
        size_t go = gbase + (size_t)t * rowstride + vcol;
        float kk = sK[t * LD + vcol], vv = sV[t * LD + vcol];
        float wkv = O[i] + tf * kk * vv;
        outp[go] = rp[go] * wkv * gp[go];
      }
    }
    __syncthreads();
  }

  const size_t OUT0 = (size_t)B_SZ * L_SZ * H_SZ * D_SZ;
  float* so = outp + OUT0 + (size_t)(b * H_SZ + h) * D_SZ * D_SZ;
  for (int i = tid; i < (D_SZ * D_SZ) / 4; i += NT) {
    int row = i >> 4, q = (i & 15) * 4;
    *(float4*)&so[row * D_SZ + q] = *(const float4*)&sS[row * LDS_S + q];
  }
}

extern "C" void kernel_launch(void* const* d_in, const int* in_sizes, int n_in,
                              void* d_out, int out_size, void* d_ws, size_t ws_size,
                              hipStream_t stream) {
  (void)in_sizes; (void)n_in; (void)out_size;
  const float* kp  = (const float*)d_in[0];
  const float* vp  = (const float*)d_in[1];
  const float* dtp = (const float*)d_in[2];
  const float* bvp = (const float*)d_in[3];
  // d_in[4] = C_vec : unused by the reference
  const float* rp  = (const float*)d_in[5];
  const float* gp  = (const float*)d_in[6];
  const float* bdp = (const float*)d_in[7];
  const float* tfp = (const float*)d_in[8];
  float* outp = (float*)d_out;

  const size_t nDS   = (size_t)BH * CH2 * (size_t)(D_SZ * D_SZ);  // 8,388,608
  const size_t nDEnd = (size_t)BH * CH2 * (size_t)D_SZ;           // 131,072
  const size_t need  = (nDS + nDEnd) * sizeof(float);             // ~34 MB

  if (ws_size >= need && d_ws != nullptr) {
    float* dSp   = (float*)d_ws;
    float* dEndp = dSp + nDS;
    p1_chunk_state<<<dim3(BH * CH2), dim3(128), 0, stream>>>(kp, vp, dtp, bvp, bdp,
                                                             dSp, dEndp);
    p2_scan<<<dim3(BH), dim3(256), 0, stream>>>(dSp, dEndp, outp);
    p3_output<<<dim3(BH * CH2), dim3(256), 0, stream>>>(kp, vp, dtp, bvp, rp, gp,
                                                        bdp, tfp, dSp, outp);
  } else {
    gla_chunk_scan_kernel<<<dim3(BH), dim3(NT), 0, stream>>>(kp, vp, dtp, bvp, rp, gp,
                                                             bdp, tfp, outp);
  }
}